// GCN_12412455486107
// MI455X (gfx1250) — compile-verified
//
#include <hip/hip_runtime.h>

#define IN_F 128
#define HID_F 128
#define OUT_F 40
#define BN_EPS 1e-5f
#define LDS_STRIDE 132          // 128 + 4: 16 consecutive rows/cols map to 16 distinct banks
#define ROW_TILES 4             // 64 rows per block per staged W panel

typedef float v2f __attribute__((ext_vector_type(2)));
typedef float v8f __attribute__((ext_vector_type(8)));

// ---------------------------------------------------------------- utilities
__global__ void init_const_kernel(float* __restrict__ p, long long n, float v) {
    long long t = (long long)blockIdx.x * blockDim.x + threadIdx.x;
    if (t < n) p[t] = v;
}

__global__ void bias_bcast_kernel(float* __restrict__ out, const float* __restrict__ b,
                                  long long total, int F) {
    long long t = (long long)blockIdx.x * blockDim.x + threadIdx.x;
    if (t < total) out[t] = b[(int)(t % F)];
}

// deg[d] starts at 1.0 (self loop); add 1 per incoming edge
__global__ void deg_kernel(const int* __restrict__ dst, float* __restrict__ deg, int E) {
    int e = blockIdx.x * blockDim.x + threadIdx.x;
    if (e < E) atomicAdd(&deg[dst[e]], 1.0f);
}

__global__ void rsqrt_kernel(float* __restrict__ p, int n) {
    int i = blockIdx.x * blockDim.x + threadIdx.x;
    if (i < n) p[i] = rsqrtf(p[i]);
}

// ------------------------------------------------------------- WMMA GEMM
// H[n x M] = X[n x 128] @ W[128 x M]   (no bias; bias folded into aggregation init)
// blockDim.x = 32 * ceil(M/16) waves; each wave owns 16 output columns.
// Each block stages the full (padded) W panel transposed in LDS once, then
// sweeps ROW_TILES row-tiles of 16 rows against it. Inner loop per K-step:
// ds_load_b64 (A frag) + ds_load_b64 (B frag) + v_wmma_f32_16x16x4_f32.
__global__ void gemm_wmma_f32(const float* __restrict__ X, const float* __restrict__ W,
                              float* __restrict__ H, int n, int M) {
    const int K = 128;
    extern __shared__ float lds[];
    const int waves   = blockDim.x >> 5;
    const int colsPad = waves << 4;              // 16 * waves  (>= M)
    float* wtile = lds;                          // [colsPad][LDS_STRIDE], transposed W
    float* atile = lds + colsPad * LDS_STRIDE;   // [16][LDS_STRIDE]

    const int tid = threadIdx.x;

    // stage W transposed: wtile[c][k] = W[k][c]; pad columns >= M with zeros
    for (int idx = tid; idx < K * colsPad; idx += blockDim.x) {
        int k = idx / colsPad;
        int c = idx - k * colsPad;               // coalesced in c
        wtile[c * LDS_STRIDE + k] = (c < M) ? W[(size_t)k * M + c] : 0.0f;
    }

    const int lane = tid & 31;
    const int wave = tid >> 5;
    const int m  = lane & 15;                    // A-frag row / B,D-frag column
    const int kh = (lane >> 4) << 1;             // half-wave K pair: 0 or 2
    const int col = (wave << 4) + m;
    const bool colOK = (col < M);
    const int rowOff = (lane >> 4) << 3;         // lanes 16-31 hold D rows r+8

    const float* wfrag = wtile + col * LDS_STRIDE + kh;   // zero-padded: no guards
    const float* afrag = atile + m * LDS_STRIDE + kh;

    for (int rt = 0; rt < ROW_TILES; ++rt) {
        const int rowBase = (blockIdx.x * ROW_TILES + rt) << 4;

        __syncthreads();                         // wtile ready / previous tile done
        for (int idx = tid; idx < 16 * K; idx += blockDim.x) {
            int r = idx >> 7;
            int k = idx & 127;
            int row = rowBase + r;
            atile[r * LDS_STRIDE + k] = (row < n) ? X[(size_t)row * K + k] : 0.0f;
        }
        __syncthreads();

        v8f acc = {};
#pragma unroll
        for (int k0 = 0; k0 < K; k0 += 4) {
            v2f a = *(const v2f*)(afrag + k0);
            v2f b = *(const v2f*)(wfrag + k0);
            acc = __builtin_amdgcn_wmma_f32_16x16x4_f32(
                /*neg_a=*/false, a, /*neg_b=*/false, b,
                /*c_mod=*/(short)0, acc, /*reuse_a=*/false, /*reuse_b=*/false);
        }

#pragma unroll
        for (int r = 0; r < 8; ++r) {
            int row = rowBase + rowOff + r;
            if (colOK && row < n) H[(size_t)row * M + col] = acc[r];
        }
    }
}

// ------------------------------------------------------------- aggregation
// out[dst] += h[src] * dinv[src]*dinv[dst] over E edges + N self loops.
// One thread per (edge, 4-feature group); float4 gather, 4 L2 float atomics.
__global__ void agg_kernel(const float* __restrict__ h, const int* __restrict__ src,
                           const int* __restrict__ dst, const float* __restrict__ dinv,
                           float* __restrict__ out, int E, int n, int F) {
    const int groups = F >> 2;
    long long t = (long long)blockIdx.x * blockDim.x + threadIdx.x;
    long long total = (long long)(E + n) * groups;
    if (t >= total) return;
    int e = (int)(t / groups);
    int g = (int)(t % groups) << 2;
    int s, d;
    if (e < E) { s = src[e]; d = dst[e]; }
    else       { s = e - E;  d = s; }            // self loop
    float w = dinv[s] * dinv[d];
    const float4 v = *reinterpret_cast<const float4*>(h + (size_t)s * F + g);
    float* o = out + (size_t)d * F + g;
    atomicAdd(o + 0, v.x * w);
    atomicAdd(o + 1, v.y * w);
    atomicAdd(o + 2, v.z * w);
    atomicAdd(o + 3, v.w * w);
}

// ------------------------------------------------------------- batchnorm
__global__ void bn_stats_kernel(const float* __restrict__ h, float* __restrict__ sums,
                                float* __restrict__ sumsq, int n, int F) {
    int f = threadIdx.x;                         // blockDim.x == F
    float s = 0.0f, q = 0.0f;
    for (int r = blockIdx.x; r < n; r += gridDim.x) {
        float v = h[(size_t)r * F + f];
        s += v;
        q += v * v;
    }
    atomicAdd(&sums[f], s);
    atomicAdd(&sumsq[f], q);
}

__global__ void bn_relu_kernel(float* __restrict__ h, const float* __restrict__ sums,
                               const float* __restrict__ sumsq,
                               const float* __restrict__ gamma, const float* __restrict__ beta,
                               int n, int F) {
    long long t = (long long)blockIdx.x * blockDim.x + threadIdx.x;
    long long total = (long long)n * F;
    if (t >= total) return;
    int f = (int)(t % F);
    float invn = 1.0f / (float)n;
    float m = sums[f] * invn;
    float v = sumsq[f] * invn - m * m;
    float val = gamma[f] * (h[t] - m) * rsqrtf(v + BN_EPS) + beta[f];
    h[t] = fmaxf(val, 0.0f);
}

// ------------------------------------------------------------- driver
extern "C" void kernel_launch(void* const* d_in, const int* in_sizes, int n_in,
                              void* d_out, int out_size, void* d_ws, size_t ws_size,
                              hipStream_t stream) {
    const float* x   = (const float*)d_in[0];
    const int*   ei  = (const int*)  d_in[1];
    const float* W0  = (const float*)d_in[2];
    const float* b0  = (const float*)d_in[3];
    const float* W1  = (const float*)d_in[4];
    const float* b1  = (const float*)d_in[5];
    const float* W2  = (const float*)d_in[6];
    const float* b2  = (const float*)d_in[7];
    const float* g0  = (const float*)d_in[8];
    const float* be0 = (const float*)d_in[9];
    const float* g1  = (const float*)d_in[10];
    const float* be1 = (const float*)d_in[11];

    const int N = in_sizes[0] / IN_F;
    const int E = in_sizes[1] / 2;
    const int* src = ei;
    const int* dst = ei + E;

    // workspace layout (floats)
    float* ws    = (float*)d_ws;
    float* dinv  = ws;                                   // N
    float* sums  = ws + N;                               // 128
    float* sumsq = sums + HID_F;                         // 128
    size_t base  = ((size_t)N + 2 * HID_F + 63) & ~(size_t)63;
    float* bufA  = ws + base;                            // N*128
    float* bufB  = bufA + (size_t)N * HID_F;             // N*128

    const int TPB = 256;
    auto blocksFor = [](long long n, int tpb) { return (unsigned)((n + tpb - 1) / tpb); };

    // ---- gcn_norm: deg (with self loops) -> dinv
    init_const_kernel<<<blocksFor(N, TPB), TPB, 0, stream>>>(dinv, N, 1.0f);
    deg_kernel<<<blocksFor(E, TPB), TPB, 0, stream>>>(dst, dinv, E);
    rsqrt_kernel<<<blocksFor(N, TPB), TPB, 0, stream>>>(dinv, N);

    const int rowsPerBlock = 16 * ROW_TILES;
    const unsigned gemmGrid = (unsigned)((N + rowsPerBlock - 1) / rowsPerBlock);
    const int wavesH = (HID_F + 15) / 16, wavesO = (OUT_F + 15) / 16;
    const size_t ldsH = (size_t)(16 * wavesH + 16) * LDS_STRIDE * sizeof(float);
    const size_t ldsO = (size_t)(16 * wavesO + 16) * LDS_STRIDE * sizeof(float);

    const long long nh = (long long)N * HID_F;
    const long long no = (long long)N * OUT_F;
    const long long aggH = (long long)(E + N) * (HID_F / 4);
    const long long aggO = (long long)(E + N) * (OUT_F / 4);

    // ---- layer 0: H = X@W0 ; agg(+b0) ; BN+ReLU
    gemm_wmma_f32<<<gemmGrid, 32 * wavesH, ldsH, stream>>>(x, W0, bufA, N, HID_F);
    bias_bcast_kernel<<<blocksFor(nh, TPB), TPB, 0, stream>>>(bufB, b0, nh, HID_F);
    agg_kernel<<<blocksFor(aggH, TPB), TPB, 0, stream>>>(bufA, src, dst, dinv, bufB, E, N, HID_F);
    init_const_kernel<<<1, 2 * HID_F, 0, stream>>>(sums, 2 * HID_F, 0.0f);
    bn_stats_kernel<<<512, HID_F, 0, stream>>>(bufB, sums, sumsq, N, HID_F);
    bn_relu_kernel<<<blocksFor(nh, TPB), TPB, 0, stream>>>(bufB, sums, sumsq, g0, be0, N, HID_F);

    // ---- layer 1
    gemm_wmma_f32<<<gemmGrid, 32 * wavesH, ldsH, stream>>>(bufB, W1, bufA, N, HID_F);
    bias_bcast_kernel<<<blocksFor(nh, TPB), TPB, 0, stream>>>(bufB, b1, nh, HID_F);
    agg_kernel<<<blocksFor(aggH, TPB), TPB, 0, stream>>>(bufA, src, dst, dinv, bufB, E, N, HID_F);
    init_const_kernel<<<1, 2 * HID_F, 0, stream>>>(sums, 2 * HID_F, 0.0f);
    bn_stats_kernel<<<512, HID_F, 0, stream>>>(bufB, sums, sumsq, N, HID_F);
    bn_relu_kernel<<<blocksFor(nh, TPB), TPB, 0, stream>>>(bufB, sums, sumsq, g1, be1, N, HID_F);

    // ---- layer 2 (output): H = h@W2 ; agg(+b2) -> d_out
    gemm_wmma_f32<<<gemmGrid, 32 * wavesO, ldsO, stream>>>(bufB, W2, bufA, N, OUT_F);
    bias_bcast_kernel<<<blocksFor(no, TPB), TPB, 0, stream>>>((float*)d_out, b2, no, OUT_F);
    agg_kernel<<<blocksFor(aggO, TPB), TPB, 0, stream>>>(bufA, src, dst, dinv, (float*)d_out, E, N, OUT_F);
}